// DeformableSelfAttention_61091614818918
// MI455X (gfx1250) — compile-verified
//
#include <hip/hip_runtime.h>

typedef float v2f __attribute__((ext_vector_type(2)));
typedef float v8f __attribute__((ext_vector_type(8)));

#define HH     200
#define WW     200
#define NQ     (HH * WW)     // 40000
#define BATCH  2
#define MTOT   (BATCH * NQ)  // 80000  (multiple of 128)
#define DDIM   256
#define NHEADS 8
#define HD     32
#define NPTS   4

// ---------------------------------------------------------------------------
// Generic fp32 WMMA GEMM:  C[M,N] = (A0 (+A1)) [M,K] @ W[K,N] + bias[N] (+resid)
// Block tile 128(M) x 64(N), K staged in chunks of 32 through LDS.
// 256 threads = 8 wave32s arranged 4(M) x 2(N); each wave owns a 32x32 C tile
// = 4 accumulators driven by V_WMMA_F32_16X16X4_F32 (exact fp32 math).
// B is staged in *fragment order* (K-pair-interleaved float2s) so every B
// fragment is a single aligned ds_load_b64 -> no VGPR packing movs.
// ---------------------------------------------------------------------------
#define BM   128
#define BN   64
#define BK   32
#define LDA  34   // padded LDS stride for A rows (even -> 8B-aligned float2 reads)
#define LDB2 66   // padded LDS stride for B pair-rows, in float2 units

__global__ __launch_bounds__(256)
void gemm_wmma_f32(const float* __restrict__ A0,
                   const float* __restrict__ A1,    // optional second A operand (added)
                   const float* __restrict__ Wt,    // [K,N]
                   const float* __restrict__ bias,  // [N]
                   const float* __restrict__ resid, // optional [M,N]
                   float* __restrict__ C,           // [M,N]
                   int M, int N, int K)
{
    __shared__ float As[BM * LDA];
    __shared__ v2f   Bs2[(BK / 2) * LDB2];   // Bs2[p][col] = (B[2p][col], B[2p+1][col])

    const int t    = threadIdx.x;
    const int lane = t & 31;            // wave32
    const int wave = t >> 5;            // 0..7
    const int wm   = (wave >> 1) * 32;  // wave M offset within block tile
    const int wn   = (wave & 1) * 32;   // wave N offset within block tile
    const int m0   = blockIdx.x * BM;
    const int n0   = blockIdx.y * BN;

    v8f acc[2][2] = {};

    const int arow = wm + (lane & 15);  // A fragment row (lanes 16-31 mirror M)
    const int half = lane >> 4;         // upper half-wave holds K=2,3 of each K4
    const int bcol = wn + (lane & 15);  // B fragment column

    for (int kc = 0; kc < K; kc += BK) {
        // ---- stage A tile (BM x BK), fusing q = x + x_pos when A1 != null ----
        #pragma unroll
        for (int i = 0; i < 4; ++i) {
            const int lin = t + i * 256;     // 0..1023 float4 slots (128 rows x 8 groups)
            const int r   = lin >> 3;
            const int kg  = lin & 7;
            const float* src0 = A0 + (size_t)(m0 + r) * K + kc + kg * 4;
            float4 a = *(const float4*)src0;
            if (A1) {
                const float4 b = *(const float4*)(A1 + (size_t)(m0 + r) * K + kc + kg * 4);
                a.x += b.x; a.y += b.y; a.z += b.z; a.w += b.w;
            }
            if (kc + BK < K) __builtin_prefetch(src0 + BK, 0, 3);  // global_prefetch_b8
            float* dst = &As[r * LDA + kg * 4];
            dst[0] = a.x; dst[1] = a.y; dst[2] = a.z; dst[3] = a.w;
        }
        // ---- stage B tile (BK x BN) in K-pair-interleaved fragment order ----
        {
            const int kp  = t >> 4;          // 0..15 : pair of K rows (2kp, 2kp+1)
            const int cg  = t & 15;          // float4 column group
            const int col = n0 + cg * 4;
            float4 ra = make_float4(0.f, 0.f, 0.f, 0.f);
            float4 rb = make_float4(0.f, 0.f, 0.f, 0.f);
            if (col + 3 < N) {               // zero-fill beyond N for narrow outputs
                ra = *(const float4*)(Wt + (size_t)(kc + 2 * kp    ) * N + col);
                rb = *(const float4*)(Wt + (size_t)(kc + 2 * kp + 1) * N + col);
            }
            float4* dst = (float4*)&Bs2[kp * LDB2 + cg * 4];
            dst[0] = make_float4(ra.x, rb.x, ra.y, rb.y);   // cols col, col+1
            dst[1] = make_float4(ra.z, rb.z, ra.w, rb.w);   // cols col+2, col+3
        }
        __syncthreads();

        // ---- 8 x K4 steps, 4 WMMAs each; all fragments are single b64 loads ----
        #pragma unroll
        for (int k4 = 0; k4 < BK; k4 += 4) {
            const int kk = k4 + half * 2;            // A: K pair within K4 block
            const int bp = (k4 >> 1) + half;         // B: fragment pair-row
            v2f a[2], b[2];
            a[0] = *(const v2f*)&As[(arow     ) * LDA + kk];
            a[1] = *(const v2f*)&As[(arow + 16) * LDA + kk];
            b[0] = Bs2[bp * LDB2 + bcol];
            b[1] = Bs2[bp * LDB2 + bcol + 16];
            #pragma unroll
            for (int mi = 0; mi < 2; ++mi)
                #pragma unroll
                for (int ni = 0; ni < 2; ++ni)
                    acc[mi][ni] = __builtin_amdgcn_wmma_f32_16x16x4_f32(
                        false, a[mi], false, b[ni], (short)0, acc[mi][ni], false, false);
        }
        __syncthreads();
    }

    // ---- epilogue: bias (+ residual), guarded store for N < block width ----
    const int rowHalf = half * 8;        // C layout: lanes 16-31 hold M = r+8
    #pragma unroll
    for (int mi = 0; mi < 2; ++mi) {
        #pragma unroll
        for (int ni = 0; ni < 2; ++ni) {
            const int col = n0 + wn + ni * 16 + (lane & 15);
            if (col < N) {
                const float bv = bias ? bias[col] : 0.f;
                #pragma unroll
                for (int r = 0; r < 8; ++r) {
                    const int row = m0 + wm + mi * 16 + rowHalf + r;
                    float val = acc[mi][ni][r] + bv;
                    if (resid) val += resid[(size_t)row * N + col];
                    C[(size_t)row * N + col] = val;
                }
            }
        }
    }
}

// ---------------------------------------------------------------------------
// Softmax over the 4 sampling points of each (query, head).
// ---------------------------------------------------------------------------
__global__ __launch_bounds__(256)
void softmax_p4(float* __restrict__ logits /* [MTOT, 32] */)
{
    const int t = blockIdx.x * blockDim.x + threadIdx.x;
    if (t >= MTOT * NHEADS) return;
    float* p = logits + (size_t)(t >> 3) * (NHEADS * NPTS) + (t & 7) * NPTS;
    const float a0 = p[0], a1 = p[1], a2 = p[2], a3 = p[3];
    const float mx = fmaxf(fmaxf(a0, a1), fmaxf(a2, a3));
    const float e0 = __expf(a0 - mx), e1 = __expf(a1 - mx);
    const float e2 = __expf(a2 - mx), e3 = __expf(a3 - mx);
    const float inv = 1.0f / (e0 + e1 + e2 + e3);
    p[0] = e0 * inv; p[1] = e1 * inv; p[2] = e2 * inv; p[3] = e3 * inv;
}

// ---------------------------------------------------------------------------
// Deformable bilinear sampling. One wave per (query m, head h); lane = channel
// (HD == 32 == wave32). px = ix + off_x, py = iy + off_y (the 0.5s cancel).
// Each corner gather is a contiguous 128B line -> full VMEM efficiency.
// ---------------------------------------------------------------------------
__global__ __launch_bounds__(256)
void msda_sample(const float* __restrict__ off,   // [MTOT, 64]  (h, p, {x,y})
                 const float* __restrict__ attn,  // [MTOT, 32]  softmaxed
                 const float* __restrict__ v,     // [MTOT, 256]
                 float* __restrict__ out)         // [MTOT, 256]
{
    const int m    = blockIdx.x;
    const int lane = threadIdx.x & 31;
    const int h    = threadIdx.x >> 5;
    const int b    = m / NQ;
    const int n    = m - b * NQ;
    const int iy   = n / WW;
    const int ix   = n - iy * WW;
    const float* vb = v + (size_t)b * NQ * DDIM + h * HD + lane;

    float acc = 0.f;
    #pragma unroll
    for (int p = 0; p < NPTS; ++p) {
        const int pi   = h * NPTS + p;
        const float ox = off[(size_t)m * 64 + pi * 2 + 0];
        const float oy = off[(size_t)m * 64 + pi * 2 + 1];
        const float aw = attn[(size_t)m * 32 + pi];
        const float px = (float)ix + ox;
        const float py = (float)iy + oy;
        const float fx = floorf(px), fy = floorf(py);
        const float lx = px - fx,   ly = py - fy;
        const int x0 = (int)fx, y0 = (int)fy;
        #pragma unroll
        for (int cy = 0; cy < 2; ++cy) {
            #pragma unroll
            for (int cx = 0; cx < 2; ++cx) {
                const int xi = x0 + cx, yi = y0 + cy;
                const float wgt = (cx ? lx : 1.f - lx) * (cy ? ly : 1.f - ly) * aw;
                const bool valid = (xi >= 0) && (xi < WW) && (yi >= 0) && (yi < HH);
                const int xc = min(max(xi, 0), WW - 1);
                const int yc = min(max(yi, 0), HH - 1);
                const float val = vb[(size_t)(yc * WW + xc) * DDIM];
                acc += valid ? wgt * val : 0.f;
            }
        }
    }
    out[(size_t)m * DDIM + h * HD + lane] = acc;
}

// ---------------------------------------------------------------------------
extern "C" void kernel_launch(void* const* d_in, const int* in_sizes, int n_in,
                              void* d_out, int out_size, void* d_ws, size_t ws_size,
                              hipStream_t stream)
{
    (void)in_sizes; (void)n_in; (void)out_size; (void)ws_size;

    const float* x      = (const float*)d_in[0];
    const float* x_pos  = (const float*)d_in[1];
    const float* W_off  = (const float*)d_in[2];
    const float* b_off  = (const float*)d_in[3];
    const float* W_attn = (const float*)d_in[4];
    const float* b_attn = (const float*)d_in[5];
    const float* W_val  = (const float*)d_in[6];
    const float* b_val  = (const float*)d_in[7];
    const float* W_out  = (const float*)d_in[8];
    const float* b_out  = (const float*)d_in[9];
    float* out = (float*)d_out;

    float* ws       = (float*)d_ws;
    float* off_buf  = ws;                               // 80000 * 64
    float* attn_buf = off_buf  + (size_t)MTOT * 64;     // 80000 * 32
    float* v_buf    = attn_buf + (size_t)MTOT * 32;     // 80000 * 256
    float* s_buf    = v_buf    + (size_t)MTOT * 256;    // 80000 * 256

    const dim3 blk(256);
    const int  mBlocks = MTOT / BM;   // 625

    // 1) offsets = (x + x_pos) @ W_off + b_off        [M, 64]
    gemm_wmma_f32<<<dim3(mBlocks, 1), blk, 0, stream>>>(
        x, x_pos, W_off, b_off, nullptr, off_buf, MTOT, NHEADS * NPTS * 2, DDIM);
    // 2) attn logits = (x + x_pos) @ W_attn + b_attn  [M, 32]
    gemm_wmma_f32<<<dim3(mBlocks, 1), blk, 0, stream>>>(
        x, x_pos, W_attn, b_attn, nullptr, attn_buf, MTOT, NHEADS * NPTS, DDIM);
    // 3) v = x @ W_val + b_val                        [M, 256]
    gemm_wmma_f32<<<dim3(mBlocks, DDIM / BN), blk, 0, stream>>>(
        x, nullptr, W_val, b_val, nullptr, v_buf, MTOT, DDIM, DDIM);
    // 4) softmax over the 4 points of each (query, head)
    softmax_p4<<<(MTOT * NHEADS + 255) / 256, blk, 0, stream>>>(attn_buf);
    // 5) deformable bilinear sampling                 [M, 256]
    msda_sample<<<MTOT, blk, 0, stream>>>(off_buf, attn_buf, v_buf, s_buf);
    // 6) out = x + sampled @ W_out + b_out            [M, 256]
    gemm_wmma_f32<<<dim3(mBlocks, DDIM / BN), blk, 0, stream>>>(
        s_buf, nullptr, W_out, b_out, x, out, MTOT, DDIM, DDIM);
}